// ProposalLayer_74320114090263
// MI455X (gfx1250) — compile-verified
//
#include <hip/hip_runtime.h>
#include <hip/hip_bf16.h>

#define BATCH 16
#define NANCH 261888
#define KPRE  6000
#define PROP  1000
#define CAP   8192          // candidate buffer per batch (power of 2 for bitonic)
#define BINS  1024
#define CHUNK 16384         // elements per histogram/compact block

typedef unsigned int  u32x4 __attribute__((ext_vector_type(4)));
typedef int           i32x8 __attribute__((ext_vector_type(8)));
typedef int           i32x4 __attribute__((ext_vector_type(4)));

// Issue a 1-row TDM DMA: n_dwords dwords from global 'ga' -> LDS offset 'lds_off'.
// Wave-level op (EXEC ignored); caller guards so only one wave issues it.
__device__ __forceinline__ void tdm_load_row(unsigned lds_off, unsigned long long ga,
                                             unsigned n_dwords) {
#if defined(__HIP_DEVICE_COMPILE__)
    u32x4 g0;
    g0[0] = 1u;                                                 // count=1, user mode
    g0[1] = lds_off;                                            // lds_addr
    g0[2] = (unsigned)(ga & 0xFFFFFFFFull);                     // global_addr[31:0]
    g0[3] = (unsigned)((ga >> 32) & 0x1FFFFFFull) | (2u << 30); // global_addr[56:32], type=2
    i32x8 g1;
    g1[0] = (int)(2u << 16);            // data_size=4B, wg_mask=0, no pad/iterate/barrier
    g1[1] = (int)((n_dwords & 0xFFFFu) << 16);   // tensor_dim0 lo16
    g1[2] = (int)((n_dwords >> 16) | (1u << 16)); // tensor_dim0 hi16 | tensor_dim1 lo16=1
    g1[3] = (int)((n_dwords & 0xFFFFu) << 16);   // tensor_dim1 hi16=0 | tile_dim0
    g1[4] = 1;                          // tile_dim1=1, tile_dim2=0
    g1[5] = (int)n_dwords;              // tensor_dim0_stride lo32
    g1[6] = 0; g1[7] = 0;
    i32x4 z4 = {0, 0, 0, 0};
    i32x8 z8 = {0, 0, 0, 0, 0, 0, 0, 0};
    __builtin_amdgcn_tensor_load_to_lds(g0, g1, z4, z4, z8, 0);
    __builtin_amdgcn_s_wait_tensorcnt(0);
#endif
}

// ---------------- K0: zero workspace counters ----------------
__global__ void k_zero(unsigned* __restrict__ hist, unsigned* __restrict__ ccnt) {
    int i = blockIdx.x * blockDim.x + threadIdx.x;
    if (i < BATCH * BINS) hist[i] = 0u;
    if (i < BATCH)        ccnt[i] = 0u;
}

// ---------------- K1: per-batch score histogram ----------------
__global__ void k_hist(const float* __restrict__ probs, unsigned* __restrict__ hist) {
    __shared__ unsigned h[BINS];
    const int b = blockIdx.y;
    for (int i = threadIdx.x; i < BINS; i += blockDim.x) h[i] = 0u;
    __syncthreads();
    const int base = blockIdx.x * CHUNK;
    const float2* p = (const float2*)(probs + (size_t)b * NANCH * 2);
    for (int i = threadIdx.x; i < CHUNK; i += blockDim.x) {
        int a = base + i;
        if (a < NANCH) {
            __builtin_prefetch(p + a + 1024, 0, 1);   // global_prefetch_b8
            float s = p[a].y;                          // global_load_b64, take channel 1
            int bin = (int)(s * (float)BINS);
            bin = bin < 0 ? 0 : (bin > BINS - 1 ? BINS - 1 : bin);
            atomicAdd(&h[bin], 1u);
        }
    }
    __syncthreads();
    for (int i = threadIdx.x; i < BINS; i += blockDim.x)
        if (h[i]) atomicAdd(&hist[b * BINS + i], h[i]);
}

// ---------------- K2: find threshold bin per batch ----------------
__global__ void k_thresh(const unsigned* __restrict__ hist, unsigned* __restrict__ tbin) {
    const int b = blockIdx.x;
    if (threadIdx.x == 0) {
        unsigned sum = 0; int t = 0;
        for (int i = BINS - 1; i >= 0; --i) {
            sum += hist[b * BINS + i];
            if (sum >= KPRE) { t = i; break; }
        }
        tbin[b] = (unsigned)t;
    }
}

// ---------------- K3: compact candidates >= threshold bin ----------------
__global__ void k_compact(const float* __restrict__ probs, const unsigned* __restrict__ tbin,
                          unsigned* __restrict__ ccnt, unsigned long long* __restrict__ cand) {
    const int b = blockIdx.y;
    const unsigned T = tbin[b];
    const int base = blockIdx.x * CHUNK;
    const float2* p = (const float2*)(probs + (size_t)b * NANCH * 2);
    for (int i = threadIdx.x; i < CHUNK; i += blockDim.x) {
        int a = base + i;
        if (a < NANCH) {
            float s = p[a].y;
            int bin = (int)(s * (float)BINS);
            bin = bin < 0 ? 0 : (bin > BINS - 1 ? BINS - 1 : bin);
            if ((unsigned)bin >= T) {
                unsigned pos = atomicAdd(&ccnt[b], 1u);
                if (pos < CAP) {
                    unsigned sb = __float_as_uint(s);
                    // key: high = score bits (positive floats order as uints),
                    //      low  = ~index so equal scores sort lower-index-first (descending)
                    cand[(size_t)b * CAP + pos] =
                        ((unsigned long long)sb << 32) | (unsigned)(~(unsigned)a);
                }
            }
        }
    }
}

// ---------------- K4: TDM-stage + bitonic sort (desc) + gather + box decode/clip ----------------
__global__ void __launch_bounds__(1024, 1)
k_sort_boxes(const unsigned long long* __restrict__ cand, const unsigned* __restrict__ ccnt,
             const float4* __restrict__ deltas, const float4* __restrict__ anchors,
             float4* __restrict__ boxes_out) {
    __shared__ unsigned long long keys[CAP];   // 64 KB — fits CDNA5 320 KB LDS easily
    const int b = blockIdx.x;
    unsigned cnt = ccnt[b]; if (cnt > CAP) cnt = CAP;

    // DMA the whole candidate buffer (64 KB = 16384 dwords) into LDS in one TDM op,
    // then zero the tail so stale entries beyond cnt can't affect the sort.
    if (threadIdx.x == 0) {
        tdm_load_row((unsigned)(size_t)(&keys[0]),
                     (unsigned long long)(size_t)(cand + (size_t)b * CAP),
                     CAP * 2u);
    }
    __syncthreads();
    for (int i = (int)cnt + (int)threadIdx.x; i < CAP; i += blockDim.x) keys[i] = 0ull;
    __syncthreads();

    for (int k = 2; k <= CAP; k <<= 1) {
        for (int j = k >> 1; j > 0; j >>= 1) {
            for (int i = threadIdx.x; i < CAP; i += blockDim.x) {
                int p = i ^ j;
                if (p > i) {
                    bool desc = ((i & k) == 0);
                    unsigned long long a = keys[i], c = keys[p];
                    if (desc ? (a < c) : (a > c)) { keys[i] = c; keys[p] = a; }
                }
            }
            __syncthreads();
        }
    }

    // top-KPRE: gather anchors/deltas (b128 loads), apply deltas * std, clip to [0,1]
    for (int t = threadIdx.x; t < KPRE; t += blockDim.x) {
        unsigned long long key = keys[t];
        unsigned idx = ~(unsigned)(key & 0xFFFFFFFFull);
        float4 o = make_float4(0.f, 0.f, 0.f, 0.f);
        if (idx < NANCH) {
            float4 av = anchors[(size_t)b * NANCH + idx];
            float4 dv = deltas [(size_t)b * NANCH + idx];
            float y1 = av.x, x1 = av.y, y2 = av.z, x2 = av.w;
            float h = y2 - y1, w = x2 - x1;
            float dy = dv.x * 0.1f, dx = dv.y * 0.1f;
            float dh = dv.z * 0.2f, dw = dv.w * 0.2f;
            float cy = y1 + 0.5f * h + dy * h;
            float cx = x1 + 0.5f * w + dx * w;
            h *= expf(dh); w *= expf(dw);
            float ny1 = cy - 0.5f * h, nx1 = cx - 0.5f * w;
            float ny2 = ny1 + h,       nx2 = nx1 + w;
            o.x = fminf(fmaxf(ny1, 0.f), 1.f);
            o.y = fminf(fmaxf(nx1, 0.f), 1.f);
            o.z = fminf(fmaxf(ny2, 0.f), 1.f);
            o.w = fminf(fmaxf(nx2, 0.f), 1.f);
        }
        boxes_out[(size_t)b * KPRE + t] = o;   // global_store_b128
    }
}

// ---------------- K5: per-batch greedy NMS, boxes staged in LDS via TDM ----------------
__global__ void __launch_bounds__(1024, 1)
k_nms(const float4* __restrict__ boxes_in, float* __restrict__ out) {
    __shared__ float4 bx[KPRE];             // 96,000 B box tile (ds_load_b128 per IoU test)
    __shared__ unsigned char supp[KPRE];    // suppression flags
    __shared__ int s_pick, s_cur;
    const int b = blockIdx.x;

    if (threadIdx.x == 0) {
        // Tensor DMA: 1 row x 24000 dwords (96,000 B) global -> LDS
        tdm_load_row((unsigned)(size_t)(&bx[0]),
                     (unsigned long long)(size_t)(boxes_in + (size_t)b * KPRE),
                     KPRE * 4u);
    }
    for (int i = threadIdx.x; i < KPRE; i += blockDim.x) supp[i] = 0;
    if (threadIdx.x == 0) s_cur = 0;
    __syncthreads();

    float* outb = out + (size_t)b * PROP * 4;
    for (int slot = 0; slot < PROP; ++slot) {
        if (threadIdx.x == 0) {
            int i = s_cur;
            while (i < KPRE && supp[i]) ++i;
            s_pick = i;
            s_cur = i + 1;
            if (i < KPRE) supp[i] = 1;
        }
        __syncthreads();
        const int i = s_pick;
        if (i >= KPRE) {   // exhausted: zero-fill remaining slots (uniform branch)
            for (int s2 = slot * 4 + (int)threadIdx.x; s2 < PROP * 4; s2 += blockDim.x)
                outb[s2] = 0.f;
            break;
        }
        const float4 bi = bx[i];
        if (threadIdx.x == 0) {
            outb[slot * 4 + 0] = bi.x; outb[slot * 4 + 1] = bi.y;
            outb[slot * 4 + 2] = bi.z; outb[slot * 4 + 3] = bi.w;
        }
        const float area = (bi.z - bi.x) * (bi.w - bi.y);
        for (int j = i + 1 + (int)threadIdx.x; j < KPRE; j += blockDim.x) {
            if (!supp[j]) {
                float4 bj = bx[j];                         // ds_load_b128
                float yy1 = fmaxf(bi.x, bj.x), xx1 = fmaxf(bi.y, bj.y);
                float yy2 = fminf(bi.z, bj.z), xx2 = fminf(bi.w, bj.w);
                float inter = fmaxf(yy2 - yy1, 0.f) * fmaxf(xx2 - xx1, 0.f);
                float areaj = (bj.z - bj.x) * (bj.w - bj.y);
                float iou = inter / (area + areaj - inter + 1e-9f);
                if (iou > 0.7f) supp[j] = 1;
            }
        }
        __syncthreads();
    }
}

// ---------------- host: launch chain ----------------
extern "C" void kernel_launch(void* const* d_in, const int* in_sizes, int n_in,
                              void* d_out, int out_size, void* d_ws, size_t ws_size,
                              hipStream_t stream) {
    const float* probs   = (const float*)d_in[0];    // (B, N, 2)
    const float4* deltasp = (const float4*)d_in[1];  // (B, N, 4)
    const float4* anchors = (const float4*)d_in[2];  // (B, N, 4)
    float* out = (float*)d_out;                      // (B, 1000, 4)

    char* ws = (char*)d_ws;
    unsigned*            hist  = (unsigned*)(ws + 0);                 // 64 KB
    unsigned*            tbin  = (unsigned*)(ws + 65536);             // 64 B
    unsigned*            ccnt  = (unsigned*)(ws + 65600);             // 64 B
    unsigned long long*  cand  = (unsigned long long*)(ws + 65664);   // 1 MB
    float4*              boxes = (float4*)(ws + 65664 + (size_t)BATCH * CAP * 8); // 1.5 MB

    const int blk_x = (NANCH + CHUNK - 1) / CHUNK;   // 16

    k_zero   <<<dim3((BATCH * BINS + 255) / 256), 256, 0, stream>>>(hist, ccnt);
    k_hist   <<<dim3(blk_x, BATCH), 256, 0, stream>>>(probs, hist);
    k_thresh <<<dim3(BATCH), 32, 0, stream>>>(hist, tbin);
    k_compact<<<dim3(blk_x, BATCH), 256, 0, stream>>>(probs, tbin, ccnt, cand);
    k_sort_boxes<<<dim3(BATCH), 1024, 0, stream>>>(cand, ccnt, deltasp, anchors, boxes);
    k_nms    <<<dim3(BATCH), 1024, 0, stream>>>(boxes, out);
}